// VectorQuantizer_18872086298730
// MI455X (gfx1250) — compile-verified
//
#include <hip/hip_runtime.h>

typedef __attribute__((ext_vector_type(2))) float v2f;
typedef __attribute__((ext_vector_type(4))) float v4f;
typedef __attribute__((ext_vector_type(8))) float v8f;

#define N_TOTAL        65536     // 16*64*64 rows
#define CH             32        // channels / code dim
#define KCODES         1024
#define ROWS_PER_BLOCK 256       // 8 waves x 32 rows (2 M-tiles per wave)
#define KCHUNK         128
#define NCHUNKS        (KCODES / KCHUNK)
// z tile stride: split-half fragment reads hit channels c and c+2 in one
// instruction -> need (2*P mod 64) in [16,48]; 2*266 mod 64 = 20.  Store phase
// (consecutive rows per lane) is conflict-free for any stride.
#define ZT_STRIDE      266
// codebook row stride (floats): 36 == 4 (mod 32) makes the b64 fragment loads
// (lane li -> bank group 4*m; halves fill offsets {0,1} vs {2,3}) touch all 64
// banks exactly once.  36*4 = 144 bytes -> rows stay 16B aligned for v4f reads.
#define CB_STRIDE      36

__global__ __launch_bounds__(256) void vq_argmin_kernel(
    const float* __restrict__ z,         // [16,32,64,64] NCHW
    const float* __restrict__ codebook,  // [1024,32]
    float* __restrict__ zq_out,          // [16,32,64,64] NCHW
    int* __restrict__ idx_out) {         // [16,64,64] int32

  __shared__ float zt[CH][ZT_STRIDE];       // z tile,   [channel][row]   34048 B
  __shared__ float cb[KCHUNK][CB_STRIDE];   // codebook, [code][dim]      18432 B
  __shared__ float c2s[KCHUNK];             // ||c_k||^2                    512 B
  __shared__ int   bks[ROWS_PER_BLOCK];     // best code per row           1024 B

  const int tid  = threadIdx.x;
  const int lane = tid & 31;
  const int wv   = tid >> 5;     // wave id 0..7 -> 32-row group
  const int li   = lane & 15;    // position within half (M for A, N for B/D)
  const int hi   = lane >> 4;    // half select (K offset for A/B, +8 rows for D)

  const int n0    = blockIdx.x * ROWS_PER_BLOCK;             // first flat row
  const int zbase = (n0 >> 12) * (CH * 4096) + (n0 & 4095);  // NCHW base (256 | 4096)

  // ---- stage z tile: 32 channels x 256 rows, coalesced 256-float runs ----
  #pragma unroll
  for (int i = 0; i < (CH * ROWS_PER_BLOCK) / 256; ++i) {
    int e = i * 256 + tid;
    int c = e >> 8;
    int r = e & 255;
    zt[c][r] = z[zbase + c * 4096 + r];
  }
  __syncthreads();

  // ---- A fragments (16x4 f32 layout), two 16-row M-tiles, loop-invariant ----
  // lane li -> M = wv*32 + t*16 + li ; half hi -> K rows {0,1} vs {2,3}
  v2f a[2][8];
  #pragma unroll
  for (int t = 0; t < 2; ++t) {
    const int mrow = wv * 32 + t * 16 + li;
    #pragma unroll
    for (int s = 0; s < 8; ++s) {
      a[t][s].x = zt[4 * s + 2 * hi][mrow];
      a[t][s].y = zt[4 * s + 2 * hi + 1][mrow];
    }
  }

  float best[2][8];
  int   bestk[2][8];
  #pragma unroll
  for (int t = 0; t < 2; ++t)
    #pragma unroll
    for (int r = 0; r < 8; ++r) { best[t][r] = __builtin_inff(); bestk[t][r] = 0; }

  for (int kc = 0; kc < NCHUNKS; ++kc) {
    __syncthreads();  // previous chunk's compute done before overwrite
    // load chunk: per wave+iter one full code, one dim per lane ->
    // coalesced global read, 32-consecutive-float LDS store (conflict-free)
    #pragma unroll
    for (int i = 0; i < (CH * KCHUNK) / 256; ++i) {
      int e = i * 256 + tid;
      int k = e >> 5;
      int d = e & 31;
      cb[k][d] = codebook[kc * (KCHUNK * CH) + e];
    }
    __syncthreads();
    // per-code squared norms via 16B LDS reads (rows are 16B aligned)
    if (tid < KCHUNK) {
      float c2 = 0.f;
      const v4f* row = (const v4f*)&cb[tid][0];
      #pragma unroll
      for (int q = 0; q < CH / 4; ++q) {
        v4f v = row[q];
        c2 = fmaf(v.x, v.x, c2);
        c2 = fmaf(v.y, v.y, c2);
        c2 = fmaf(v.z, v.z, c2);
        c2 = fmaf(v.w, v.w, c2);
      }
      c2s[tid] = c2;
    }
    __syncthreads();

    // ---- 8 k-tiles of 16 codes; each b64 B-fragment feeds two WMMAs ----
    for (int kt = 0; kt < KCHUNK / 16; ++kt) {
      const int k0    = kt * 16;
      const float c2k = c2s[k0 + li];
      v8f acc0 = {};
      v8f acc1 = {};
      #pragma unroll
      for (int s = 0; s < 8; ++s) {
        // B 4x16: lanes hold N=li; half hi selects K rows {0,1}/{2,3}
        const v2f b = *(const v2f*)&cb[k0 + li][4 * s + 2 * hi];
        acc0 = __builtin_amdgcn_wmma_f32_16x16x4_f32(
            false, a[0][s], false, b, (short)0, acc0, false, false);
        acc1 = __builtin_amdgcn_wmma_f32_16x16x4_f32(
            false, a[1][s], false, b, (short)0, acc1, false, false);
      }
      // score = ||c||^2 - 2*dot ; ascending k + strict < keeps first min
      const int kg = kc * KCHUNK + k0 + li;
      #pragma unroll
      for (int r = 0; r < 8; ++r) {
        float s0 = fmaf(-2.f, acc0[r], c2k);
        if (s0 < best[0][r]) { best[0][r] = s0; bestk[0][r] = kg; }
        float s1 = fmaf(-2.f, acc1[r], c2k);
        if (s1 < best[1][r]) { best[1][r] = s1; bestk[1][r] = kg; }
      }
    }
  }

  // ---- argmin across the 16 lanes of each half (k candidates across lanes) ----
  #pragma unroll
  for (int t = 0; t < 2; ++t) {
    #pragma unroll
    for (int r = 0; r < 8; ++r) {
      float s = best[t][r];
      int   k = bestk[t][r];
      #pragma unroll
      for (int m = 1; m <= 8; m <<= 1) {
        float so = __shfl_xor(s, m, 32);
        int   ko = __shfl_xor(k, m, 32);
        if (so < s || (so == s && ko < k)) { s = so; k = ko; }
      }
      const int lrow = wv * 32 + t * 16 + r + 8 * hi;  // D: VGPR r -> M=r / M=r+8
      if (li == 0) {
        bks[lrow] = k;
        idx_out[n0 + lrow] = k;
      }
    }
  }
  __syncthreads();

  // ---- gather codes -> z_q (NCHW), coalesced per-channel runs of 256 ----
  #pragma unroll
  for (int i = 0; i < (CH * ROWS_PER_BLOCK) / 256; ++i) {
    int e = i * 256 + tid;
    int c = e >> 8;
    int r = e & 255;
    zq_out[zbase + c * 4096 + r] = codebook[bks[r] * CH + c];
  }
}

extern "C" void kernel_launch(void* const* d_in, const int* in_sizes, int n_in,
                              void* d_out, int out_size, void* d_ws, size_t ws_size,
                              hipStream_t stream) {
  (void)in_sizes; (void)n_in; (void)out_size; (void)d_ws; (void)ws_size;
  const float* z        = (const float*)d_in[0];
  const float* codebook = (const float*)d_in[1];
  float* zq  = (float*)d_out;                          // 16*32*64*64 floats
  int*   idx = (int*)((float*)d_out + N_TOTAL * CH);   // 16*64*64 int32, appended
  vq_argmin_kernel<<<dim3(N_TOTAL / ROWS_PER_BLOCK), dim3(256), 0, stream>>>(
      z, codebook, zq, idx);
}